// DecoderTimeAttention_31696858644745
// MI455X (gfx1250) — compile-verified
//
#include <hip/hip_runtime.h>

typedef float v2f __attribute__((ext_vector_type(2)));
typedef float v8f __attribute__((ext_vector_type(8)));

constexpr int TL  = 64;    // sequence length per tile
constexpr int TD  = 128;   // model dim
constexpr int TH  = 8;     // heads
constexpr int ST  = 132;   // padded LDS row stride for 128-wide buffers (bank-conflict free)
constexpr int SST = 68;    // padded LDS row stride for 64-wide score buffers

__device__ __forceinline__ v8f wmma4(v2f a, v2f b, v8f c) {
  // D = A(16x4,f32) * B(4x16,f32) + C(16x16,f32)
  return __builtin_amdgcn_wmma_f32_16x16x4_f32(false, a, false, b, (short)0, c,
                                               false, false);
}

// C[64 x 16] (column block w) = A[64 x 128] (LDS, stride ST) @ W[128 x 128] (global)
//   + bias (+ optional residual from LDS), result -> dst (LDS, stride ST)
__device__ __forceinline__ void gemm_proj(const float* __restrict__ Asrc,
                                          const float* __restrict__ Wg,
                                          const float* __restrict__ bias,
                                          float* __restrict__ dst,
                                          const float* resid,
                                          int w, int lr, int half) {
  const int col = (w << 4) + lr;
  v8f acc[4] = {};
  for (int ks = 0; ks < 32; ++ks) {
    const int k0 = (ks << 2) + (half << 1);
    v2f b;
    b.x = Wg[k0 * TD + col];
    b.y = Wg[(k0 + 1) * TD + col];
#pragma unroll
    for (int m = 0; m < 4; ++m) {
      const float* ap = Asrc + (16 * m + lr) * ST + k0;
      v2f a; a.x = ap[0]; a.y = ap[1];
      acc[m] = wmma4(a, b, acc[m]);
    }
  }
  const float bv = bias[col];
#pragma unroll
  for (int m = 0; m < 4; ++m) {
#pragma unroll
    for (int j = 0; j < 8; ++j) {
      const int row = 16 * m + 8 * half + j;
      float x = acc[m][j] + bv;
      if (resid) x += resid[row * ST + col];
      dst[row * ST + col] = x;
    }
  }
}

__global__ __launch_bounds__(256)
void decoder_time_attn_kernel(const float* __restrict__ Qg, const float* __restrict__ Kg,
                              const float* __restrict__ Tg,
                              const float* __restrict__ Wq, const float* __restrict__ bq,
                              const float* __restrict__ Wk, const float* __restrict__ bk,
                              const float* __restrict__ Wv, const float* __restrict__ bv,
                              const float* __restrict__ Wot, const float* __restrict__ bot,
                              const float* __restrict__ Wtg, const float* __restrict__ btg,
                              const float* __restrict__ g_time, const float* __restrict__ b_time,
                              const float* __restrict__ g_tgt, const float* __restrict__ b_tgt,
                              float* __restrict__ attn_out, float* __restrict__ tval_out,
                              float* __restrict__ ctgt_out) {
  extern __shared__ float lds[];
  float* sQ  = lds;            // raw Q tile (kept for residual)        64*ST
  float* sTV = sQ + TL * ST;   // raw K tile, later merged TV           64*ST
  float* sA  = sTV + TL * ST;  // q projection, later raw Tgt tile      64*ST
  float* sB  = sA + TL * ST;   // k projection, later merged CT         64*ST
  float* sV  = sB + TL * ST;   // v projection                          64*ST
  float* sS  = sV + TL * ST;   // per-head scores, later O1/O2          8*64*SST
  float* sO1 = sS;             // pre-LN time branch   (64*ST)
  float* sO2 = sS + TL * ST;   // pre-LN target branch (64*ST)

  const int tid  = threadIdx.x;
  const int lane = tid & 31;
  const int w    = tid >> 5;   // wave id == head id == column block id
  const int lr   = lane & 15;
  const int half = lane >> 4;
  const size_t bn = blockIdx.x;
  const size_t tilebase = bn * (size_t)(TL * TD);

  // ---- Phase 0: stage Q and K tiles (coalesced float4) ----
  {
    const float4* q4 = (const float4*)(Qg + tilebase);
    const float4* k4 = (const float4*)(Kg + tilebase);
    for (int i = tid; i < TL * TD / 4; i += 256) {
      const int e = i << 2, r = e >> 7, c = e & (TD - 1);
      *(float4*)&sQ[r * ST + c]  = q4[i];
      *(float4*)&sTV[r * ST + c] = k4[i];
    }
  }
  __syncthreads();

  // ---- Phase 1: q / k / v projections ----
  gemm_proj(sQ,  Wq, bq, sA, nullptr, w, lr, half);
  gemm_proj(sTV, Wk, bk, sB, nullptr, w, lr, half);
  gemm_proj(sTV, Wv, bv, sV, nullptr, w, lr, half);
  __syncthreads();

  // ---- Phase 2: per-head scores S_h = q_h @ k_h^T / 4  (wave w = head w) ----
  {
    float* Sh = sS + w * (TL * SST);
    float* attn_h = attn_out + (bn * TH + w) * (size_t)(TL * TL);
    for (int m = 0; m < 4; ++m) {
      v8f acc[4] = {};
      for (int ks = 0; ks < 4; ++ks) {
        const int k0 = (w << 4) + (ks << 2) + (half << 1);
        v2f a;
        a.x = sA[(16 * m + lr) * ST + k0];
        a.y = sA[(16 * m + lr) * ST + k0 + 1];
#pragma unroll
        for (int nt = 0; nt < 4; ++nt) {
          v2f b;  // B[d][j] = k[j][d]  -> A-style fragment from k projection
          b.x = sB[(16 * nt + lr) * ST + k0];
          b.y = sB[(16 * nt + lr) * ST + k0 + 1];
          acc[nt] = wmma4(a, b, acc[nt]);
        }
      }
#pragma unroll
      for (int nt = 0; nt < 4; ++nt) {
#pragma unroll
        for (int j = 0; j < 8; ++j) {
          const int row = 16 * m + 8 * half + j;
          const int col = 16 * nt + lr;
          const float s = acc[nt][j] * 0.25f;  // 1/sqrt(16)
          Sh[row * SST + col] = s;
          attn_h[row * TL + col] = s;
        }
      }
    }
  }
  __syncthreads();

  // ---- Phase 3a: stage Tgt tile into sA (q is dead) ----
  {
    const float4* t4 = (const float4*)(Tg + tilebase);
    for (int i = tid; i < TL * TD / 4; i += 256) {
      const int e = i << 2, r = e >> 7, c = e & (TD - 1);
      *(float4*)&sA[r * ST + c] = t4[i];
    }
  }

  // ---- Phase 3b: TV columns for head w:  tv_h = S_h @ v_h ----
  {
    const float* Sh = sS + w * (TL * SST);
    v8f acc[4] = {};
    for (int ks = 0; ks < 16; ++ks) {
      const int k0 = (ks << 2) + (half << 1);
      v2f b;
      b.x = sV[k0 * ST + (w << 4) + lr];
      b.y = sV[(k0 + 1) * ST + (w << 4) + lr];
#pragma unroll
      for (int m = 0; m < 4; ++m) {
        v2f a;
        a.x = Sh[(16 * m + lr) * SST + k0];
        a.y = Sh[(16 * m + lr) * SST + k0 + 1];
        acc[m] = wmma4(a, b, acc[m]);
      }
    }
#pragma unroll
    for (int m = 0; m < 4; ++m)
#pragma unroll
      for (int j = 0; j < 8; ++j)
        sTV[(16 * m + 8 * half + j) * ST + (w << 4) + lr] = acc[m][j];
  }
  __syncthreads();  // Tgt staged + TV complete

  // ---- Phase 3c: CT columns for head w:  ct_h = S_h @ tgt_h ----
  {
    const float* Sh = sS + w * (TL * SST);
    v8f acc[4] = {};
    for (int ks = 0; ks < 16; ++ks) {
      const int k0 = (ks << 2) + (half << 1);
      v2f b;
      b.x = sA[k0 * ST + (w << 4) + lr];
      b.y = sA[(k0 + 1) * ST + (w << 4) + lr];
#pragma unroll
      for (int m = 0; m < 4; ++m) {
        v2f a;
        a.x = Sh[(16 * m + lr) * SST + k0];
        a.y = Sh[(16 * m + lr) * SST + k0 + 1];
        acc[m] = wmma4(a, b, acc[m]);
      }
    }
#pragma unroll
    for (int m = 0; m < 4; ++m)
#pragma unroll
      for (int j = 0; j < 8; ++j)
        sB[(16 * m + 8 * half + j) * ST + (w << 4) + lr] = acc[m][j];
  }
  __syncthreads();  // CT complete; scores dead -> sO1/sO2 reuse

  // ---- Phase 4: output projections (O1 gets +bias +residual Q, O2 gets +bias) ----
  gemm_proj(sTV, Wot, bot, sO1, sQ,      w, lr, half);
  gemm_proj(sB,  Wtg, btg, sO2, nullptr, w, lr, half);
  __syncthreads();

  // ---- Phase 5: LayerNorm, one row per thread (128 rows total) ----
  if (tid < 2 * TL) {
    const int row = tid & (TL - 1);
    const bool isT = tid < TL;
    const float* src = (isT ? sO1 : sO2) + row * ST;
    const float* gg = isT ? g_time : g_tgt;
    const float* bb = isT ? b_time : b_tgt;
    float* gout = (isT ? tval_out : ctgt_out) + (bn * (size_t)TL + row) * TD;
    float mu = 0.f, s2 = 0.f;
    for (int c = 0; c < TD; ++c) { const float x = src[c]; mu += x; s2 += x * x; }
    mu *= (1.0f / TD);
    const float var = s2 * (1.0f / TD) - mu * mu;
    const float rs = rsqrtf(var + 1e-5f);
    for (int c = 0; c < TD; ++c)
      gout[c] = (src[c] - mu) * rs * gg[c] + bb[c];
  }
}

extern "C" void kernel_launch(void* const* d_in, const int* in_sizes, int n_in,
                              void* d_out, int out_size, void* d_ws, size_t ws_size,
                              hipStream_t stream) {
  const float* Qg  = (const float*)d_in[0];
  const float* Kg  = (const float*)d_in[1];
  const float* Tg  = (const float*)d_in[2];
  const float* Wq  = (const float*)d_in[3];
  const float* bq  = (const float*)d_in[4];
  const float* Wk  = (const float*)d_in[5];
  const float* bk  = (const float*)d_in[6];
  const float* Wv  = (const float*)d_in[7];
  const float* bv  = (const float*)d_in[8];
  const float* Wot = (const float*)d_in[9];
  const float* bot = (const float*)d_in[10];
  const float* Wtg = (const float*)d_in[11];
  const float* btg = (const float*)d_in[12];
  const float* g_time = (const float*)d_in[13];
  const float* b_time = (const float*)d_in[14];
  const float* g_tgt  = (const float*)d_in[15];
  const float* b_tgt  = (const float*)d_in[16];

  const int BN = in_sizes[0] / (TL * TD);  // B*N = 828
  float* attn = (float*)d_out;
  float* tval = attn + (size_t)BN * TH * TL * TL;
  float* ctgt = tval + (size_t)BN * TL * TD;

  const size_t shmem = (size_t)(5 * TL * ST + TH * TL * SST) * sizeof(float);
  decoder_time_attn_kernel<<<BN, 256, shmem, stream>>>(
      Qg, Kg, Tg, Wq, bq, Wk, bk, Wv, bv, Wot, bot, Wtg, btg,
      g_time, b_time, g_tgt, b_tgt, attn, tval, ctgt);
}